// FAC_ACLDN_44925357916858
// MI455X (gfx1250) — compile-verified
//
#include <hip/hip_runtime.h>
#include <hip/hip_bf16.h>
#include <math.h>
#include <stdint.h>

typedef __attribute__((ext_vector_type(2))) float v2f;
typedef __attribute__((ext_vector_type(8))) float v8f;

#define kB   64
#define kH   256
#define kW   256
#define kHW  (kH * kW)           // 65536
#define kN   ((size_t)kB * kHW)  // 4194304
#define kOH  86
#define kOW  86
#define kSP  (kOH * kOW)         // 7396
#define kTsh 0.1f                // theta * lambda, theta = 1
#define kEps 1e-5f

// ---------------------------------------------------------------------------
// helpers
// ---------------------------------------------------------------------------
// branchless zero-padded load: clamp address (always valid), cndmask the value
__device__ __forceinline__ float ldc256(const float* p, int y, int x) {
  int yc = min(max(y, 0), kH - 1), xc = min(max(x, 0), kW - 1);
  float v = p[(yc << 8) + xc];
  return (y == yc && x == xc) ? v : 0.f;
}

__device__ __forceinline__ v8f wmma4(v2f a, v2f b, v8f c) {
  // V_WMMA_F32_16X16X4_F32 : D = A(16x4) * B(4x16) + C(16x16), all f32
  return __builtin_amdgcn_wmma_f32_16x16x4_f32(false, a, false, b, (short)0, c,
                                               false, false);
}

// LDS byte offset of a __shared__ object: low 32 bits of the flat pointer
// (LDS aperture: ADDR[31:0] is the LDS byte address, ISA 10.2)
__device__ __forceinline__ unsigned lds_off(const void* p) {
  return (unsigned)(uintptr_t)p;
}

// async global -> LDS copies (ASYNCcnt-tracked, no VGPR staging)
__device__ __forceinline__ void async_ld_b32(unsigned lds, const float* g) {
  asm volatile("global_load_async_to_lds_b32 %0, %1, off" :: "v"(lds), "v"(g) : "memory");
}
__device__ __forceinline__ void async_ld_b128(unsigned lds, const float* g) {
  asm volatile("global_load_async_to_lds_b128 %0, %1, off" :: "v"(lds), "v"(g) : "memory");
}
__device__ __forceinline__ void wait_async0() {
  asm volatile("s_wait_asynccnt 0x0" ::: "memory");
}

// ---------------------------------------------------------------------------
// global min / max reduction over x
// ---------------------------------------------------------------------------
__global__ void fac_minmax1(const float* __restrict__ x, float* __restrict__ part, int n) {
  __shared__ float smn[256], smx[256];
  int tid = threadIdx.x;
  float mn = 3.4e38f, mx = -3.4e38f;
  for (int i = blockIdx.x * 256 + tid; i < n; i += gridDim.x * 256) {
    float v = x[i];
    mn = fminf(mn, v);
    mx = fmaxf(mx, v);
  }
  smn[tid] = mn; smx[tid] = mx;
  __syncthreads();
  for (int s = 128; s > 0; s >>= 1) {
    if (tid < s) {
      smn[tid] = fminf(smn[tid], smn[tid + s]);
      smx[tid] = fmaxf(smx[tid], smx[tid + s]);
    }
    __syncthreads();
  }
  if (tid == 0) { part[blockIdx.x] = smn[0]; part[512 + blockIdx.x] = smx[0]; }
}

__global__ void fac_minmax2(const float* __restrict__ part, float* __restrict__ mm) {
  __shared__ float smn[256], smx[256];
  int tid = threadIdx.x;
  smn[tid] = fminf(part[tid], part[tid + 256]);
  smx[tid] = fmaxf(part[512 + tid], part[768 + tid]);
  __syncthreads();
  for (int s = 128; s > 0; s >>= 1) {
    if (tid < s) {
      smn[tid] = fminf(smn[tid], smn[tid + s]);
      smx[tid] = fmaxf(smx[tid], smx[tid + s]);
    }
    __syncthreads();
  }
  if (tid == 0) { mm[0] = smn[0]; mm[1] = smx[0]; }
}

// f = (x - min) / (max - min + 1e-8)
__global__ void fac_norm(const float* __restrict__ x, const float* __restrict__ mm,
                         float* __restrict__ f) {
  int i = blockIdx.x * 256 + threadIdx.x;
  float mn = mm[0];
  float sc = 1.0f / (mm[1] - mn + 1e-8f);
  f[i] = (x[i] - mn) * sc;
}

// ---------------------------------------------------------------------------
// edge map g + AC state init (p=0, v=0, r^0 = -f since theta=1)
// ---------------------------------------------------------------------------
__global__ void fac_g_init(const float* __restrict__ f, float* __restrict__ g,
                           float* __restrict__ r, float* __restrict__ p1,
                           float* __restrict__ p2, float* __restrict__ v) {
  int i = blockIdx.x * 256 + threadIdx.x;
  int b = i >> 16, p = i & 0xFFFF, y = p >> 8, x = p & 255;
  const float* fb = f + (size_t)b * kHW;
  float a00 = ldc256(fb, y - 1, x - 1), a01 = ldc256(fb, y - 1, x), a02 = ldc256(fb, y - 1, x + 1);
  float a10 = ldc256(fb, y, x - 1),                                  a12 = ldc256(fb, y, x + 1);
  float a20 = ldc256(fb, y + 1, x - 1), a21 = ldc256(fb, y + 1, x), a22 = ldc256(fb, y + 1, x + 1);
  float gx = -a00 + a02 - 2.f * a10 + 2.f * a12 - a20 + a22;
  float gy = -a00 - 2.f * a01 - a02 + a20 + 2.f * a21 + a22;
  g[i] = 1.0f / (1.0f + gx * gx + gy * gy);   // beta = 1
  r[i] = -fb[p];                              // r^0 = -(f - 0)/theta
  p1[i] = 0.f; p2[i] = 0.f; v[i] = 0.f;
}

// ---------------------------------------------------------------------------
// B step with LDS-tiled r: block = 64x4 pixel tile, halo-1 staged in LDS.
// Interior tiles stage via global_load_async_to_lds_b32 (ASYNCcnt path);
// border tiles use the zero-padded scalar path.
// ggx/ggy = Sobel(r)/8 ; p = (p + 0.125*gg) / (1 + 0.125*|gg|/g)
// ---------------------------------------------------------------------------
__global__ void fac_B(const float* __restrict__ r, const float* __restrict__ g,
                      float* __restrict__ p1, float* __restrict__ p2) {
  __shared__ float rt[6 * 66];
  int tid = threadIdx.x;
  int tx = blockIdx.x & 3;          // 4 x-tiles of 64
  int ty = (blockIdx.x >> 2) & 63;  // 64 y-tiles of 4
  int b  = blockIdx.x >> 8;
  const float* rb = r + (size_t)b * kHW;
  bool interior = (ty >= 1) && (ty <= 62) && (tx >= 1) && (tx <= 2);  // block-uniform
  if (interior) {
    unsigned base = lds_off(rt);
    for (int e = tid; e < 396; e += 256) {
      int row = e / 66, col = e - row * 66;
      int gy = ty * 4 - 1 + row, gx = tx * 64 - 1 + col;
      async_ld_b32(base + e * 4, rb + (gy << 8) + gx);
    }
    wait_async0();
  } else {
    for (int e = tid; e < 396; e += 256) {
      int row = e / 66, col = e - row * 66;
      rt[e] = ldc256(rb, ty * 4 - 1 + row, tx * 64 - 1 + col);
    }
  }
  __syncthreads();
  int lx = tid & 63, ly = tid >> 6;
  int y = ty * 4 + ly, x = tx * 64 + lx;
  int i = (b << 16) | (y << 8) | x;
  __builtin_prefetch(p1 + i, 0, 1);           // global_prefetch_b8
  __builtin_prefetch(p2 + i, 0, 1);
  const float* c0 = &rt[ly * 66 + lx];        // 3x3 window top-left
  float a00 = c0[0],   a01 = c0[1],   a02 = c0[2];
  float a10 = c0[66],                 a12 = c0[68];
  float a20 = c0[132], a21 = c0[133], a22 = c0[134];
  float ggx = 0.125f * (-a00 + a02 - 2.f * a10 + 2.f * a12 - a20 + a22);
  float ggy = 0.125f * (-a00 - 2.f * a01 - a02 + a20 + 2.f * a21 + a22);
  float denom = 1.0f + 0.125f * sqrtf(ggx * ggx + ggy * ggy) / g[i];
  p1[i] = (p1[i] + 0.125f * ggx) / denom;
  p2[i] = (p2[i] + 0.125f * ggy) / denom;
}

// ---------------------------------------------------------------------------
// A step (fused): D = div(p); vt = v + D; v' = shrink(vt); r = D - f + v'
// ---------------------------------------------------------------------------
__global__ void fac_A(const float* __restrict__ f, const float* __restrict__ p1,
                      const float* __restrict__ p2, float* __restrict__ v,
                      float* __restrict__ r) {
  int i = blockIdx.x * 256 + threadIdx.x;
  int b = i >> 16, p = i & 0xFFFF, y = p >> 8, x = p & 255;
  const float* p1b = p1 + (size_t)b * kHW;
  const float* p2b = p2 + (size_t)b * kHW;
  float D = ldc256(p1b, y, x + 1) - ldc256(p1b, y, x - 1)
          + ldc256(p2b, y + 1, x) - ldc256(p2b, y - 1, x);
  float vt = v[i] + D;
  float vn = (vt > kTsh) ? vt - kTsh : ((vt < -kTsh) ? vt + kTsh : 0.f);
  v[i] = vn;
  r[i] = D - f[i] + vn;
}

// final iteration: u = f - v - div(p); binary = u > 0.5
__global__ void fac_Afin(const float* __restrict__ f, const float* __restrict__ p1,
                         const float* __restrict__ p2, const float* __restrict__ v,
                         float* __restrict__ bin) {
  int i = blockIdx.x * 256 + threadIdx.x;
  int b = i >> 16, p = i & 0xFFFF, y = p >> 8, x = p & 255;
  const float* p1b = p1 + (size_t)b * kHW;
  const float* p2b = p2 + (size_t)b * kHW;
  float D = ldc256(p1b, y, x + 1) - ldc256(p1b, y, x - 1)
          + ldc256(p2b, y + 1, x) - ldc256(p2b, y - 1, x);
  float u = f[i] - v[i] - D;
  bin[i] = (u > 0.5f) ? 1.f : 0.f;
}

// ---------------------------------------------------------------------------
// head: conv1+bn+relu -> dw(s3) -> pw -> bn+relu ; writes h2 in HWC layout
// ---------------------------------------------------------------------------
__global__ void fac_head(const float* __restrict__ bin,
                         const float* __restrict__ w1, const float* __restrict__ cb1,
                         const float* __restrict__ s1, const float* __restrict__ bb1,
                         const float* __restrict__ m1, const float* __restrict__ v1,
                         const float* __restrict__ dww, const float* __restrict__ pww,
                         const float* __restrict__ sdw, const float* __restrict__ bdw,
                         const float* __restrict__ mdw, const float* __restrict__ vdw,
                         float* __restrict__ h2) {
  int t = blockIdx.x * 256 + threadIdx.x;   // exactly 64*7396 threads
  int b = t / kSP, sp = t % kSP;
  int oy = sp / kOW, ox = sp % kOW;
  const float* bp = bin + (size_t)b * kHW;
  float patch[5][5];
  int iy0 = 3 * oy - 2, ix0 = 3 * ox - 2;
#pragma unroll
  for (int dy = 0; dy < 5; ++dy)
#pragma unroll
    for (int dx = 0; dx < 5; ++dx)
      patch[dy][dx] = ldc256(bp, iy0 + dy, ix0 + dx);
  float accdw[16];
#pragma unroll
  for (int c = 0; c < 16; ++c) {
    float sc = rsqrtf(v1[c] + kEps) * s1[c];
    float sh = bb1[c] + (cb1[c] - m1[c]) * sc;   // bn(conv + bias)
    float wl[9];
#pragma unroll
    for (int q = 0; q < 9; ++q) wl[q] = w1[c * 9 + q];
    float acc = 0.f;
#pragma unroll
    for (int ky = 0; ky < 3; ++ky)
#pragma unroll
      for (int kx = 0; kx < 3; ++kx) {
        int iy = 3 * oy + ky - 1, ix = 3 * ox + kx - 1;
        bool pad = (iy < 0 || iy >= kH || ix < 0 || ix >= kW);   // dw zero-pad
        float cv = 0.f;
#pragma unroll
        for (int u = 0; u < 3; ++u)
#pragma unroll
          for (int w = 0; w < 3; ++w) cv += patch[ky + u][kx + w] * wl[u * 3 + w];
        float a = fmaxf(cv * sc + sh, 0.f);
        acc += pad ? 0.f : a * dww[c * 9 + ky * 3 + kx];
      }
    accdw[c] = acc;
  }
  float outc[16];
#pragma unroll
  for (int co = 0; co < 16; ++co) {
    float o = 0.f;
#pragma unroll
    for (int ci = 0; ci < 16; ++ci) o += pww[co * 16 + ci] * accdw[ci];
    float sc = rsqrtf(vdw[co] + kEps) * sdw[co];
    outc[co] = fmaxf(o * sc + (bdw[co] - mdw[co] * sc), 0.f);
  }
  // HWC store: 16 contiguous floats per pixel -> 4x b128
  float* dst = h2 + ((size_t)b * kSP + sp) * 16;
#pragma unroll
  for (int q = 0; q < 4; ++q) {
    float4 vv = {outc[q * 4], outc[q * 4 + 1], outc[q * 4 + 2], outc[q * 4 + 3]};
    *(float4*)(dst + q * 4) = vv;
  }
}

// pack conv3 weights: w3p[s][co][kk] = w3[co][cb*4+kk][t], s = t*4+cb
__global__ void fac_pack3(const float* __restrict__ w3, float* __restrict__ w3p) {
  int i = blockIdx.x * 256 + threadIdx.x;      // 9216 exact (36 blocks)
  int kk = i & 3, co = (i >> 2) & 63, s = i >> 8;
  int t = s >> 2, cb = s & 3;
  w3p[i] = w3[(co * 16 + cb * 4 + kk) * 9 + t];
}

// ---------------------------------------------------------------------------
// conv3 (3x3, 16->64, pad 1) + bn3 + relu, implicit GEMM via WMMA f32 16x16x4.
// HWC in/out. Block = (b, y, 32-col pair); 8 waves = 4 co-tiles x 2 n-tiles
// sharing a 3x34x16 LDS activation tile. B frags = ds_load_b64, A = b64.
// ---------------------------------------------------------------------------
__global__ void fac_conv3(const float* __restrict__ h2,
                          const float* __restrict__ w3p, const float* __restrict__ b3,
                          const float* __restrict__ s3, const float* __restrict__ bb3,
                          const float* __restrict__ m3, const float* __restrict__ v3,
                          float* __restrict__ h3) {
  __shared__ float tile[3 * 34 * 16];          // 6528 B
  int tid = threadIdx.x;
  int xt2 = blockIdx.x % 3;                    // 3 pairs of 32 output cols
  int y   = (blockIdx.x / 3) % kOH;
  int b   = blockIdx.x / (3 * kOH);
  const float* src = h2 + (size_t)b * kSP * 16;

  // cooperative staging: (j=row 0..2, cc=col 0..33, c4=ch-quad 0..3) -> float4
  for (int e4 = tid; e4 < 3 * 34 * 4; e4 += 256) {   // 408 float4 loads
    int c4 = e4 & 3;
    int cc = (e4 >> 2) % 34;
    int j  = (e4 >> 2) / 34;
    int iy = y - 1 + j;
    int ix = xt2 * 32 - 1 + cc;
    float4 vv = {0.f, 0.f, 0.f, 0.f};
    if (iy >= 0 && iy < kOH && ix >= 0 && ix < kOW)
      vv = *(const float4*)(src + ((size_t)iy * kOW + ix) * 16 + c4 * 4);
    *(float4*)(tile + e4 * 4) = vv;
  }
  __syncthreads();

  int wv  = tid >> 5;
  int lane = tid & 31;
  int ct  = wv & 3;                            // co tile
  int xtl = wv >> 2;                           // n tile within 32-col pair
  int n   = lane & 15;
  int hi  = lane >> 4;                         // K-pair: 0 -> K0,K1 ; 1 -> K2,K3
  int aco = ct * 16 + n;                       // A-fragment row (out channel)
  v8f acc = {0.f, 0.f, 0.f, 0.f, 0.f, 0.f, 0.f, 0.f};
#pragma unroll
  for (int t = 0; t < 9; ++t) {
    int ky = t / 3, kx = t % 3;
    int cc = xtl * 16 + n + kx;                // LDS col 0..33
    const float* brow = tile + (ky * 34 + cc) * 16;
#pragma unroll
    for (int cb = 0; cb < 4; ++cb) {
      int s = t * 4 + cb;
      v2f a  = *(const v2f*)(w3p + s * 256 + aco * 4 + hi * 2);   // b64
      v2f bf = *(const v2f*)(brow + cb * 4 + hi * 2);             // ds b64
      acc = wmma4(a, bf, acc);
    }
  }
  int xcol = xt2 * 32 + xtl * 16 + n;
  if (xcol < kOW) {
    float o[8];
#pragma unroll
    for (int j = 0; j < 8; ++j) {
      int co = ct * 16 + hi * 8 + j;
      float inv = rsqrtf(v3[co] + kEps) * s3[co];
      o[j] = fmaxf((acc[j] + b3[co] - m3[co]) * inv + bb3[co], 0.f);
    }
    float* dst = h3 + ((size_t)b * kSP + y * kOW + xcol) * 64 + ct * 16 + hi * 8;
    float4 lo = {o[0], o[1], o[2], o[3]};
    float4 hi4 = {o[4], o[5], o[6], o[7]};
    *(float4*)(dst)     = lo;                  // b128
    *(float4*)(dst + 4) = hi4;                 // b128
  }
}

// ---------------------------------------------------------------------------
// conv4 (1x1, 64->128) + bias + bn4 + relu + global-average-pool, fused.
// HWC input. Block = (b, nt); 16x64 activation tile staged in LDS via
// global_load_async_to_lds_b128 (clamped addresses; OOB columns are masked
// after the WMMA at the reduction). Reused by 8 mt-waves.
// ---------------------------------------------------------------------------
__global__ void fac_conv4gap(const float* __restrict__ h3,
                             const float* __restrict__ w4, const float* __restrict__ b4,
                             const float* __restrict__ s4, const float* __restrict__ bb4,
                             const float* __restrict__ m4, const float* __restrict__ v4,
                             float* __restrict__ pooled) {
  __shared__ float tile[16 * 64];              // 4 KB
  int tid = threadIdx.x;
  int nt = blockIdx.x % 463;                   // ceil(7396/16)
  int b  = blockIdx.x / 463;
  const float* src = h3 + (size_t)b * kSP * 64;
  {                                            // one async b128 per thread
    int lsp = tid >> 4, c4 = tid & 15;
    int sp = nt * 16 + lsp;
    if (sp >= kSP) sp = kSP - 1;               // clamp; column masked later
    async_ld_b128(lds_off(tile) + tid * 16, src + (size_t)sp * 64 + c4 * 4);
    wait_async0();
  }
  __syncthreads();

  int mt = tid >> 5;                           // 8 co-tiles of 16
  int lane = tid & 31;
  int n  = lane & 15;
  int hi = lane >> 4;
  int sp = nt * 16 + n;
  bool inb = sp < kSP;
  int aco = mt * 16 + n;
  const float* brow = tile + n * 64;
  v8f acc = {0.f, 0.f, 0.f, 0.f, 0.f, 0.f, 0.f, 0.f};
#pragma unroll
  for (int cb = 0; cb < 16; ++cb) {
    int kb = cb * 4 + hi * 2;
    v2f a  = *(const v2f*)(w4 + aco * 64 + kb);    // b64, contiguous ci
    v2f bf = *(const v2f*)(brow + kb);             // ds b64
    acc = wmma4(a, bf, acc);
  }
  const float wsc = 1.0f / (float)kSP;
#pragma unroll
  for (int j = 0; j < 8; ++j) {
    int co = mt * 16 + hi * 8 + j;
    float inv = rsqrtf(v4[co] + kEps) * s4[co];
    float val = (acc[j] + b4[co] - m4[co]) * inv + bb4[co];
    val = fmaxf(val, 0.f);
    if (!inb) val = 0.f;                       // mask OOB columns
    val += __shfl_xor(val, 1, 32);             // reduce 16 columns in half-wave
    val += __shfl_xor(val, 2, 32);
    val += __shfl_xor(val, 4, 32);
    val += __shfl_xor(val, 8, 32);
    if (n == 0) atomicAdd(&pooled[b * 128 + co], val * wsc);
  }
}

__global__ void fac_zero(float* __restrict__ p, int n) {
  int i = blockIdx.x * 256 + threadIdx.x;
  if (i < n) p[i] = 0.f;
}

__global__ void fac_fc(const float* __restrict__ pooled, const float* __restrict__ fcw,
                       const float* __restrict__ fcb, float* __restrict__ out) {
  int t = blockIdx.x * 256 + threadIdx.x;
  if (t >= 64 * 38) return;
  int b = t / 38, o = t % 38;
  const float* pb = pooled + b * 128;
  const float* wr = fcw + o * 128;
  float s = fcb[o];
#pragma unroll 8
  for (int k = 0; k < 128; ++k) s += pb[k] * wr[k];
  out[t] = s;
}

// ---------------------------------------------------------------------------
extern "C" void kernel_launch(void* const* d_in, const int* in_sizes, int n_in,
                              void* d_out, int out_size, void* d_ws, size_t ws_size,
                              hipStream_t stream) {
  (void)in_sizes; (void)n_in; (void)out_size; (void)ws_size;
  const float* x   = (const float*)d_in[0];
  const float* w1  = (const float*)d_in[1];
  const float* cb1 = (const float*)d_in[2];
  const float* s1  = (const float*)d_in[3];
  const float* bb1 = (const float*)d_in[4];
  const float* m1  = (const float*)d_in[5];
  const float* v1  = (const float*)d_in[6];
  const float* dww = (const float*)d_in[7];
  const float* pww = (const float*)d_in[8];
  const float* sdw = (const float*)d_in[9];
  const float* bdw = (const float*)d_in[10];
  const float* mdw = (const float*)d_in[11];
  const float* vdw = (const float*)d_in[12];
  const float* w3  = (const float*)d_in[13];
  const float* b3  = (const float*)d_in[14];
  const float* s3  = (const float*)d_in[15];
  const float* bb3 = (const float*)d_in[16];
  const float* m3  = (const float*)d_in[17];
  const float* v3  = (const float*)d_in[18];
  const float* w4  = (const float*)d_in[19];
  const float* b4  = (const float*)d_in[20];
  const float* s4  = (const float*)d_in[21];
  const float* bb4 = (const float*)d_in[22];
  const float* m4  = (const float*)d_in[23];
  const float* v4  = (const float*)d_in[24];
  const float* fcw = (const float*)d_in[25];
  const float* fcb = (const float*)d_in[26];
  float* out = (float*)d_out;

  // workspace layout (floats); classifier aliases dead AC buffers
  float* W   = (float*)d_ws;
  float* f   = W;                    // [N]
  float* g   = W + kN;               // [N]
  float* p1  = W + 2 * kN;           // [N]
  float* p2  = W + 3 * kN;           // [N]
  float* v   = W + 4 * kN;           // [N]
  float* r   = W + 5 * kN;           // [N]
  float* bin = g;                    // alias: g dead after last B step
  float* h2  = W + 2 * kN;           // alias p1/p2: 64*7396*16 = 7,573,504 < 2N
  float* h3  = W + 4 * kN;           // alias v/r + tail: 64*7396*64 = 30,294,016
  float* pooled = W + 4 * kN + 30294016; // [8192]
  float* part   = pooled + 8192;         // [1024]
  float* mm     = part + 1024;           // [2]
  float* w3p    = mm + 2;                // [9216] packed conv3 weights

  const int EW = (int)(kN / 256);    // 16384 blocks, exact

  fac_minmax1<<<512, 256, 0, stream>>>(x, part, (int)kN);
  fac_minmax2<<<1, 256, 0, stream>>>(part, mm);
  fac_norm<<<EW, 256, 0, stream>>>(x, mm, f);
  fac_g_init<<<EW, 256, 0, stream>>>(f, g, r, p1, p2, v);

  for (int it = 0; it < 100; ++it) {
    fac_B<<<EW, 256, 0, stream>>>(r, g, p1, p2);
    if (it < 99)
      fac_A<<<EW, 256, 0, stream>>>(f, p1, p2, v, r);
    else
      fac_Afin<<<EW, 256, 0, stream>>>(f, p1, p2, v, bin);
  }

  fac_head<<<1849, 256, 0, stream>>>(bin, w1, cb1, s1, bb1, m1, v1,
                                     dww, pww, sdw, bdw, mdw, vdw, h2);
  fac_pack3<<<36, 256, 0, stream>>>(w3, w3p);
  fac_conv3<<<16512, 256, 0, stream>>>(h2, w3p, b3, s3, bb3, m3, v3, h3); // 64*86*3 blocks
  fac_zero<<<32, 256, 0, stream>>>(pooled, 8192);
  fac_conv4gap<<<29632, 256, 0, stream>>>(h3, w4, b4, s4, bb4, m4, v4, pooled); // 64*463
  fac_fc<<<10, 256, 0, stream>>>(pooled, fcw, fcb, out);
}